// kernel_generated_1_21887153341321
// MI455X (gfx1250) — compile-verified
//
#include <hip/hip_runtime.h>

// CDNA5 (gfx1250) grouped-conv pipeline as two fp32 WMMA GEMMs.
// Stage 1: C1(8x3136)  = w1(8x192)  x im2col_H(x)   -> t4 in d_ws (100352 B)
// Stage 2: C2(32x6272) = w2(32x12)  x im2col_W(t4)  -> d_out
// ~14.5 MFLOP / ~1.7 MB total => latency-bound on MI455X (23.3 TB/s, 192MB L2);
// keep fp32 (V_WMMA_F32_16X16X4_F32), branch-free inner loops, immediate-offset
// loads, intermediate stays L2-resident between the two passes.
//
// K-permutation trick: GEMM is K-order invariant if A and B agree, so K is
// reordered (tap-major) to make the tap index a compile-time constant in the
// fully-unrolled loops -> no integer division, no EXEC-masked loads.

typedef __attribute__((ext_vector_type(2))) float v2f;
typedef __attribute__((ext_vector_type(8))) float v8f;

static __device__ __forceinline__ v8f wmma_f32(v2f a, v2f b, v8f c) {
  return __builtin_amdgcn_wmma_f32_16x16x4_f32(
      /*neg_a=*/false, a, /*neg_b=*/false, b,
      /*c_mod=*/(short)0, c, /*reuse_a=*/false, /*reuse_b=*/false);
}

// ---------------------------------------------------------------------------
// Stage 1: M=8 (padded to 16), K=192 (order k' = j*64 + i), N=3136.
// One wave per 16-column tile.  A'[m,k'] = w1[m*192 + i*3 + j].
// B'[k',col(o,m,n)] = mask_j * x[(i*4+o)*784 + ((m+j+26)%28)*28 + n]
//   (roll-by-1 along H folded into the row index; mask_j kills the padded taps;
//    the rolled row index is always in [0,27] so loads are unconditional).
// ---------------------------------------------------------------------------
__global__ void __launch_bounds__(128)
conv_stage1_wmma(const float* __restrict__ x,
                 const float* __restrict__ w1,
                 float* __restrict__ t4) {
  const int lane = threadIdx.x & 31;
  const int wave = threadIdx.x >> 5;
  const int tile = blockIdx.x * 4 + wave;      // 0..195 (196*16 == 3136 exact)
  const int nl   = lane & 15;                  // N within tile / A row M
  const int half = lane >> 4;                  // K sub-slot select

  const int col = tile * 16 + nl;              // column index -> (o,m,n)
  const int o   = col / 784;                   // group 0..3
  const int rem = col - o * 784;
  const int m   = rem / 28;                    // H
  const int n   = rem - m * 28;                // W

  // A base: rows 8..15 alias rows 0..7 (their D rows are never stored).
  const float* ap = w1 + (nl & 7) * 192 + half * 6;   // + i0*3 + j, i0 = 4kk+2half

  v8f c0 = {}, c1 = {};                        // two chains to break WMMA RAW dep
#pragma unroll
  for (int j = 0; j < 3; ++j) {
    const int   h     = m + j - 1;             // logical padded-H tap
    const float bmask = (h >= 0 && h < 28) ? 1.0f : 0.0f;
    const int   row   = (m + j + 26) % 28;     // rolled source row, always valid
    const float* bp   = x + o * 784 + row * 28 + n + half * 2 * 3136;
#pragma unroll
    for (int kk = 0; kk < 16; ++kk) {          // i0 = kk*4 (+2*half via bases)
      v2f a, b;
      a.x = ap[(j * 64 + kk * 4) * 3 + j - j * 192 + j];  // see note below
      a.x = ap[kk * 12 + j];                   // w1[row, i0*3 + j]
      a.y = ap[kk * 12 + 3 + j];               // w1[row, (i0+1)*3 + j]
      b.x = bmask * bp[kk * 4 * 3136];         // x[ch = i0*4+o]
      b.y = bmask * bp[kk * 4 * 3136 + 3136];  // x[ch = (i0+1)*4+o]
      if (kk & 1) c1 = wmma_f32(a, b, c1);
      else        c0 = wmma_f32(a, b, c0);
    }
  }

  // D layout: VGPR r, lanes 0-15 hold M=r; only rows 0..7 are real output.
  if (half == 0) {
#pragma unroll
    for (int r = 0; r < 8; ++r)
      t4[r * 3136 + col] = c0[r] + c1[r];
  }
}

// ---------------------------------------------------------------------------
// Stage 2: M=32 (2 tiles), K=12 (order k'' = t*4 + j), N=6272.
// One wave per 16-column tile.  A''[i2,k''] = w2[i2*12 + j*3 + t].
// B''[k'',col(o,m,n)] = mask_t * t4[o*3136 + j*784 + m*28 + clamp(n+t-1)].
// ---------------------------------------------------------------------------
__global__ void __launch_bounds__(128)
conv_stage2_wmma(const float* __restrict__ t4,
                 const float* __restrict__ w2,
                 float* __restrict__ out) {
  const int lane = threadIdx.x & 31;
  const int wave = threadIdx.x >> 5;
  const int tile = blockIdx.x * 4 + wave;      // 0..391 (392*16 == 6272 exact)
  const int nl   = lane & 15;
  const int half = lane >> 4;

  const int col = tile * 16 + nl;
  const int o   = col / 784;                   // stage-1 output channel 0..7
  const int rem = col - o * 784;
  const int m   = rem / 28;
  const int n   = rem - m * 28;

  // Lane base pointers; all loop loads use compile-time immediate offsets.
  const float* ap = w2 + nl * 12 + half * 6;               // + {0,3} + t ; +192 for tile 1
  const float* bp = t4 + o * 3136 + half * 1568 + m * 28;  // + {0,784} + clamped W

  // Per-tap W clamp + 0/1 masks (branch-free; clamped index always in-bounds).
  const int   wm1   = (n > 0) ? n - 1 : 0;
  const int   wp1   = (n < 27) ? n + 1 : 27;
  const float mask0 = (n > 0) ? 1.0f : 0.0f;
  const float mask2 = (n < 27) ? 1.0f : 0.0f;
  const int   widx[3]  = { wm1, n, wp1 };
  const float wmask[3] = { mask0, 1.0f, mask2 };

  v8f c0 = {}, c1 = {};
#pragma unroll
  for (int t = 0; t < 3; ++t) {                // K step: k'' = t*4 + (2*half + {0,1})
    v2f b;
    b.x = wmask[t] * bp[widx[t]];              // group j = 2*half
    b.y = wmask[t] * bp[784 + widx[t]];        // group j = 2*half + 1
    v2f a0, a1;
    a0.x = ap[t];                              // w2[nl,      j*3 + t]
    a0.y = ap[3 + t];
    a1.x = ap[192 + t];                        // w2[16 + nl, j*3 + t]
    a1.y = ap[195 + t];
    c0 = wmma_f32(a0, b, c0);
    c1 = wmma_f32(a1, b, c1);
  }

  // out channel = i2*8 + o ; i2 = Mtile*16 + r + 8*half ; spatial = rem.
#pragma unroll
  for (int r = 0; r < 8; ++r) {
    const int i2 = r + 8 * half;
    out[(i2 * 8 + o) * 784 + rem]        = c0[r];
    out[((i2 + 16) * 8 + o) * 784 + rem] = c1[r];
  }
}

extern "C" void kernel_launch(void* const* d_in, const int* in_sizes, int n_in,
                              void* d_out, int out_size, void* d_ws, size_t ws_size,
                              hipStream_t stream) {
  const float* x  = (const float*)d_in[0];   // (1,256,28,28) fp32
  const float* w1 = (const float*)d_in[1];   // (8,64,3)  fp32
  const float* w2 = (const float*)d_in[2];   // (32,4,3)  fp32
  float* out = (float*)d_out;                // (1,256,28,28) fp32
  float* t4  = (float*)d_ws;                 // needs 25088 floats = 100352 B

  conv_stage1_wmma<<<49, 128, 0, stream>>>(x, w1, t4);   // 196 wave-tiles
  conv_stage2_wmma<<<98, 128, 0, stream>>>(t4, w2, out); // 392 wave-tiles
}